// AutoRegMLAttention_82205674045544
// MI455X (gfx1250) — compile-verified
//
#include <hip/hip_runtime.h>
#include <hip/hip_bf16.h>

// MLA decode (DeepSeek-style) for MI455X / gfx1250, wave32, f32 WMMA 16x16x4.
// S==1 => RoPE is identity (pos=0). Value path uses absorption: (P@cKV)@W_UV.
// Flash kernel double-buffers KV tiles with global_load_async_to_lds_b128.

typedef __attribute__((ext_vector_type(2))) float v2f;
typedef __attribute__((ext_vector_type(8))) float v8f;

#define NCHUNK 16
#define LT 16
#define GEMM_WAVES 8

__device__ __forceinline__ v8f wmma_f32(v2f a, v2f b, v8f c) {
    return __builtin_amdgcn_wmma_f32_16x16x4_f32(false, a, false, b, (short)0, c,
                                                 false, false);
}

// Async global->LDS copy, 16B per lane, tracked by ASYNCcnt.
__device__ __forceinline__ void async_copy_b128(unsigned lds_off, const float* g) {
    asm volatile("global_load_async_to_lds_b128 %0, %1, off"
                 :: "v"(lds_off), "v"(g) : "memory");
}

// ---------------------------------------------------------------------------
// Generic C[16,N] = A[16,K] @ B[K,N], B addressed as B[k*sk + n*sn].
// grid.x = N/16 tiles, grid.y = per-head offset multiplier.
// 8 waves split K (compile-time K so the loop unrolls), reduce through LDS.
// ---------------------------------------------------------------------------
template <int K>
__global__ __launch_bounds__(256) void gemm16_wmma(
    const float* __restrict__ A, int lda, long long aHead,
    const float* __restrict__ B, long long sk, long long sn, long long bHead,
    float* __restrict__ C, int ldc, long long cHead)
{
    const int head = blockIdx.y;
    A += (long long)head * aHead;
    B += (long long)head * bHead;
    C += (long long)head * cHead;

    const int n0   = blockIdx.x * 16;
    const int lane = threadIdx.x & 31;
    const int w    = threadIdx.x >> 5;
    const int m    = lane & 15;    // A-row (M) for A-frag, B-col (N) for B-frag
    const int hi   = lane >> 4;    // lane half selects K phase

    constexpr int kslice = K / GEMM_WAVES;      // all K here are multiples of 32
    const int kbeg = w * kslice;

    v8f acc = {0.f,0.f,0.f,0.f,0.f,0.f,0.f,0.f};
#pragma unroll 4
    for (int ko = 0; ko < kslice; ko += 4) {
        const int kk = kbeg + ko;
        v2f a, bf;
        // 16x4 f32 A: VGPR0 = K (hi?2:0), VGPR1 = K (hi?3:1), row = lane&15
        a.x = A[(long long)m * lda + kk + 2 * hi];
        a.y = A[(long long)m * lda + kk + 2 * hi + 1];
        // 4x16 f32 B: VGPR0 = row K (hi?2:0), col = lane&15
        bf.x = B[(long long)(kk + 2 * hi)     * sk + (long long)(n0 + m) * sn];
        bf.y = B[(long long)(kk + 2 * hi + 1) * sk + (long long)(n0 + m) * sn];
        acc = wmma_f32(a, bf, acc);
    }

    __shared__ float s_red[GEMM_WAVES][256];
#pragma unroll
    for (int i = 0; i < 8; ++i)
        s_red[w][(i + hi * 8) * 16 + m] = acc[i];
    __syncthreads();

    const int t = threadIdx.x;            // 256 outputs: row = t>>4, col = t&15
    float sum = 0.f;
#pragma unroll
    for (int wv = 0; wv < GEMM_WAVES; ++wv) sum += s_red[wv][t];
    C[(long long)(t >> 4) * ldc + n0 + (t & 15)] = sum;
}

// ---------------------------------------------------------------------------
// out[b][l][D] = l<L ? cached[b][l][D] : newrow[b][D]   (builds concat cache)
// ---------------------------------------------------------------------------
__global__ void build_cache(const float* __restrict__ cached,
                            const float* __restrict__ newrow,
                            float* __restrict__ out, int L, int D)
{
    const long long rowElems = (long long)(L + 1) * D;
    const long long total4 = 16LL * rowElems / 4;
    for (long long i4 = (long long)blockIdx.x * blockDim.x + threadIdx.x;
         i4 < total4; i4 += (long long)gridDim.x * blockDim.x) {
        long long e = i4 * 4;
        int b = (int)(e / rowElems);
        long long rem = e - (long long)b * rowElems;
        int l = (int)(rem / D);
        int d = (int)(rem - (long long)l * D);
        const float4* src = (l < L)
            ? (const float4*)(cached + ((long long)b * L + l) * D + d)
            : (const float4*)(newrow + (long long)b * D + d);
        *(float4*)(out + e) = *src;
    }
}

// ---------------------------------------------------------------------------
// Flash-decoding partials: grid(NCHUNK, B), 256 threads (8 waves).
// Double-buffered async staging of the KV stream; score-WMMA for all
// 32 heads, online softmax, ctx-WMMA accumulation (ctx = P @ cKV, 32x512).
// ---------------------------------------------------------------------------
__global__ __launch_bounds__(256) void mla_flash_partial(
    const float* __restrict__ qce,  // [B][32][512] effective queries
    const float* __restrict__ qr,   // [B][32][64]  rope queries
    const float* __restrict__ kv,   // [B][4097][512] concat cKV cache
    const float* __restrict__ kr,   // [B][4097][64]  concat kR cache
    float* __restrict__ part)       // [B][NCHUNK][64 + 32*512]
{
    const int chunk = blockIdx.x;
    const int b     = blockIdx.y;
    const int tid   = threadIdx.x;
    const int lane  = tid & 31;
    const int w     = tid >> 5;
    const int m     = lane & 15;
    const int hi    = lane >> 4;

    __shared__ float s_kv[2][LT][516];   // pad 516: 516%64==4 -> conflict-free
    __shared__ float s_kr[2][LT][68];
    __shared__ float s_sc[32][17];
    __shared__ float s_p[32][17];
    __shared__ float s_m[32], s_s[32], s_scale[32];

    if (tid < 32) { s_m[tid] = -3.0e38f; s_s[tid] = 0.f; }

    const v8f zacc = {0.f,0.f,0.f,0.f,0.f,0.f,0.f,0.f};
    v8f acc[8];                 // wave w owns ctx tiles g=w*8..w*8+7
#pragma unroll
    for (int t = 0; t < 8; ++t) acc[t] = zacc;

    const int Lstart = chunk * 256;
    const int Lend   = (chunk == NCHUNK - 1) ? 4097 : Lstart + 256;
    const float scl  = 0.0721687836487032f;   // 1/sqrt(HD+RH) = 1/sqrt(192)

    // Issue one tile's async copies: 8 (kv) + 1 (kr) instructions per wave.
    // Rows past the cache end are clamped to row 4096 (real finite data);
    // masking happens at score time, so no zero-fill is needed.
    auto issue_tile = [&](int bufn, int L0) {
        for (int s = tid; s < LT * 128; s += 256) {      // 16B segments of kv
            int r = s >> 7, c = (s & 127) << 2;
            int row = min(L0 + r, 4096);
            async_copy_b128((unsigned)(unsigned long long)&s_kv[bufn][r][c],
                            kv + ((long long)b * 4097 + row) * 512 + c);
        }
        for (int s = tid; s < LT * 16; s += 256) {       // 16B segments of kr
            int r = s >> 4, c = (s & 15) << 2;
            int row = min(L0 + r, 4096);
            async_copy_b128((unsigned)(unsigned long long)&s_kr[bufn][r][c],
                            kr + ((long long)b * 4097 + row) * 64 + c);
        }
    };

    issue_tile(0, Lstart);
    int buf = 0;

    for (int L0 = Lstart; L0 < Lend; L0 += LT) {
        const int nvalid = min(LT, Lend - L0);
        const bool has_next = (L0 + LT) < Lend;

        if (has_next) {
            issue_tile(buf ^ 1, L0 + LT);     // prefetch next tile (9 per wave)
            asm volatile("s_wait_asynccnt 0x9" ::: "memory");   // current done
        } else {
            asm volatile("s_wait_asynccnt 0x0" ::: "memory");
        }
        __syncthreads();

        // ---- scores: waves 0..1, one 16head x 16L tile each ----
        if (w < 2) {
            const float* Aq = qce + ((long long)b * 32 + w * 16) * 512;
            const float* Ar = qr  + ((long long)b * 32 + w * 16) * 64;
            v8f sc = zacc;
            for (int kk = 0; kk < 512; kk += 4) {          // content part
                v2f a, bf;
                a.x  = Aq[m * 512 + kk + 2 * hi];
                a.y  = Aq[m * 512 + kk + 2 * hi + 1];
                bf.x = s_kv[buf][m][kk + 2 * hi];          // B[k][L] = kv[L][k]
                bf.y = s_kv[buf][m][kk + 2 * hi + 1];
                sc = wmma_f32(a, bf, sc);
            }
            for (int kk = 0; kk < 64; kk += 4) {           // rope part
                v2f a, bf;
                a.x  = Ar[m * 64 + kk + 2 * hi];
                a.y  = Ar[m * 64 + kk + 2 * hi + 1];
                bf.x = s_kr[buf][m][kk + 2 * hi];
                bf.y = s_kr[buf][m][kk + 2 * hi + 1];
                sc = wmma_f32(a, bf, sc);
            }
#pragma unroll
            for (int i = 0; i < 8; ++i) {
                int h = w * 16 + i + hi * 8;
                s_sc[h][m] = (m < nvalid) ? sc[i] * scl : -3.0e38f;
            }
        }
        __syncthreads();

        // ---- online softmax update (one lane per head) ----
        if (tid < 32) {
            int h = tid;
            float tmax = -3.0e38f;
            for (int j = 0; j < LT; ++j) tmax = fmaxf(tmax, s_sc[h][j]);
            float newm = fmaxf(s_m[h], tmax);
            float r = __expf(s_m[h] - newm);
            float sum = 0.f;
            for (int j = 0; j < LT; ++j) {
                float p = __expf(s_sc[h][j] - newm);
                s_p[h][j] = p;
                sum += p;
            }
            s_s[h]     = s_s[h] * r + sum;
            s_m[h]     = newm;
            s_scale[h] = r;
        }
        __syncthreads();

        // ---- ctx += P(16x16) @ cKV_tile(16x512): 64 tiles over 8 waves ----
#pragma unroll
        for (int t = 0; t < 8; ++t) {
            int g  = w * 8 + t;
            int hg = g >> 5;       // head-tile 0/1
            int kt = g & 31;       // k output tile
#pragma unroll
            for (int i = 0; i < 8; ++i)
                acc[t][i] *= s_scale[hg * 16 + i + hi * 8];
#pragma unroll
            for (int kk = 0; kk < LT; kk += 4) {
                v2f a, bf;
                a.x  = s_p[hg * 16 + m][kk + 2 * hi];
                a.y  = s_p[hg * 16 + m][kk + 2 * hi + 1];
                bf.x = s_kv[buf][kk + 2 * hi][kt * 16 + m];
                bf.y = s_kv[buf][kk + 2 * hi + 1][kt * 16 + m];
                acc[t] = wmma_f32(a, bf, acc[t]);
            }
        }
        __syncthreads();   // buffers recycled next iteration only after this
        buf ^= 1;
    }

    // ---- write chunk partials: [m(32) | s(32) | ctx 32x512] ----
    float* pb = part + ((long long)b * NCHUNK + chunk) * (64 + 32 * 512);
    if (tid < 32) { pb[tid] = s_m[tid]; pb[32 + tid] = s_s[tid]; }
    float* pc = pb + 64;
#pragma unroll
    for (int t = 0; t < 8; ++t) {
        int g = w * 8 + t, hg = g >> 5, kt = g & 31;
#pragma unroll
        for (int i = 0; i < 8; ++i) {
            int h = hg * 16 + i + hi * 8;
            pc[h * 512 + kt * 16 + m] = acc[t][i];
        }
    }
}

// ---------------------------------------------------------------------------
// Combine chunk partials -> normalized ctx[b][h][512]. grid(32,16), 128 thr.
// ---------------------------------------------------------------------------
__global__ __launch_bounds__(128) void mla_reduce(
    const float* __restrict__ part, float* __restrict__ ctx)
{
    const int h = blockIdx.x;
    const int b = blockIdx.y;
    const int tid = threadIdx.x;
    const long long stride = 64 + 32 * 512;
    const float* pb = part + (long long)b * NCHUNK * stride;

    float M = -3.0e38f;
    for (int c = 0; c < NCHUNK; ++c) M = fmaxf(M, pb[c * stride + h]);
    float wgt[NCHUNK];
    float S = 0.f;
    for (int c = 0; c < NCHUNK; ++c) {
        float wv = __expf(pb[c * stride + h] - M);
        wgt[c] = wv;
        S += pb[c * stride + 32 + h] * wv;
    }
    float invS = 1.0f / S;
    for (int k = tid; k < 512; k += 128) {
        float a = 0.f;
        for (int c = 0; c < NCHUNK; ++c)
            a += wgt[c] * pb[c * stride + 64 + h * 512 + k];
        ctx[((long long)b * 32 + h) * 512 + k] = a * invS;
    }
}

// ---------------------------------------------------------------------------
extern "C" void kernel_launch(void* const* d_in, const int* in_sizes, int n_in,
                              void* d_out, int out_size, void* d_ws, size_t ws_size,
                              hipStream_t stream) {
    const float* hs     = (const float*)d_in[0];   // [16,1,4096]
    const float* ckv0   = (const float*)d_in[1];   // [16,4096,512]
    const float* ckr0   = (const float*)d_in[2];   // [16,4096,64]
    const float* W_DQ   = (const float*)d_in[3];   // [4096,1536]
    const float* W_DKV  = (const float*)d_in[4];   // [4096,512]
    const float* W_UQ_C = (const float*)d_in[5];   // [1536,4096]
    const float* W_UQ_R = (const float*)d_in[6];   // [1536,2048]
    const float* W_KR   = (const float*)d_in[7];   // [4096,64]
    const float* W_UK_C = (const float*)d_in[8];   // [512,4096]
    const float* W_UV_C = (const float*)d_in[9];   // [512,4096]
    const float* W_O    = (const float*)d_in[10];  // [4096,4096]

    float* out  = (float*)d_out;
    float* out0 = out;                               // [16,1,4096]
    float* out1 = out + 65536;                       // [16,4097,512]
    float* out2 = out + 65536 + 33562624LL;          // [16,4097,64]

    float* ws   = (float*)d_ws;
    float* cQ    = ws;            // 16x1536
    float* cKVt  = ws + 24576;    // 16x512
    float* kRt   = ws + 32768;    // 16x64
    float* qC    = ws + 33792;    // 16x4096
    float* qR    = ws + 99328;    // 16x2048
    float* qce   = ws + 132096;   // 16x32x512
    float* partb = ws + 394240;   // 16x16x(64+16384)
    float* ctx   = ws + 4604928;  // 16x32x512
    float* attn  = ws + 4867072;  // 16x4096

    dim3 blk(256);

    // Down-projections from hidden states (M=16, K=4096)
    gemm16_wmma<4096><<<dim3(96, 1), blk, 0, stream>>>(hs, 4096, 0, W_DQ,  1536, 1, 0, cQ,   1536, 0);
    gemm16_wmma<4096><<<dim3(32, 1), blk, 0, stream>>>(hs, 4096, 0, W_DKV,  512, 1, 0, cKVt,  512, 0);
    gemm16_wmma<4096><<<dim3( 4, 1), blk, 0, stream>>>(hs, 4096, 0, W_KR,    64, 1, 0, kRt,    64, 0);

    // Query up-projections (K=1536); RoPE is identity for S==1 (pos=0)
    gemm16_wmma<1536><<<dim3(256, 1), blk, 0, stream>>>(cQ, 1536, 0, W_UQ_C, 4096, 1, 0, qC, 4096, 0);
    gemm16_wmma<1536><<<dim3(128, 1), blk, 0, stream>>>(cQ, 1536, 0, W_UQ_R, 2048, 1, 0, qR, 2048, 0);

    // Absorption: qce[b,n,k] = sum_h qC[b,n*128+h] * W_UK_C[k, n*128+h]
    // per head: A=qC(+n*128, lda 4096), B: sk=1 (h), sn=4096 (k), +n*128
    gemm16_wmma<128><<<dim3(32, 32), blk, 0, stream>>>(qC, 4096, 128,
                                                       W_UK_C, 1, 4096, 128,
                                                       qce, 16384, 512);

    // Build concatenated caches directly in the output buffers
    build_cache<<<4096, 256, 0, stream>>>(ckv0, cKVt, out1, 4096, 512);
    build_cache<<<1024, 256, 0, stream>>>(ckr0, kRt,  out2, 4096, 64);

    // Flash-decoding over L=4097 (16 chunks x 16 batches)
    mla_flash_partial<<<dim3(NCHUNK, 16), 256, 0, stream>>>(qce, qR, out1, out2, partb);
    mla_reduce<<<dim3(32, 16), 128, 0, stream>>>(partb, ctx);

    // Value up-projection per head: attn[b, n*128+hd] = ctx[b,n,:] @ W_UV[:, n*128+hd]
    gemm16_wmma<512><<<dim3(8, 32), blk, 0, stream>>>(ctx, 16384, 512,
                                                      W_UV_C, 4096, 1, 128,
                                                      attn, 4096, 128);

    // Output projection: out0 = attn @ W_O  (16x4096 @ 4096x4096)
    gemm16_wmma<4096><<<dim3(256, 1), blk, 0, stream>>>(attn, 4096, 0,
                                                        W_O, 4096, 1, 0,
                                                        out0, 4096, 0);
}